// region_pooling_74878459838574
// MI455X (gfx1250) — compile-verified
//
#include <hip/hip_runtime.h>
#include <hip/hip_bf16.h>
#include <stdint.h>

// ---------------------------------------------------------------------------
// Region pooling: bilinear gather of 4x 4KB feature rows per point, weighted
// mean over 512 points per region. Memory-bound gather (L2-resident: 128MB
// feature map < 192MB L2). CDNA5 paths: global_load_async_to_lds_b128 for
// coord staging (ASYNCcnt), global_prefetch for upcoming gather rows.
// ---------------------------------------------------------------------------

typedef int v4i __attribute__((ext_vector_type(4)));

#if defined(__has_builtin)
#  if __has_builtin(__builtin_amdgcn_global_load_async_to_lds_b128)
#    define HAVE_ASYNC_LDS 1
#  endif
#  if __has_builtin(__builtin_amdgcn_s_wait_asynccnt)
#    define HAVE_WAIT_ASYNC 1
#  endif
#endif

static constexpr int Bn     = 8;
static constexpr int HWn    = 4096;   // 64 x 64
static constexpr int Cn     = 1024;
static constexpr int Mn     = 32;
static constexpr int Pn     = 512;
static constexpr int CSPLIT = 4;              // channel tiles per region
static constexpr int BLOCK  = 64;             // threads per block (2 waves)
static constexpr int CT     = Cn / CSPLIT;    // 256 channels per block
static constexpr int ROW4   = Cn / 4;         // row stride in float4 units
static constexpr int PFD    = 8;              // prefetch distance in points

struct Samp {
  int r00, r01, r10, r11;
  float w00, w01, w10, w11;
};

__device__ __forceinline__ Samp make_samp(float2 pc) {
  // reference: y = coord0*(h-1), x = coord1*(w-1), h = w = 64
  float y = pc.x * 63.0f;
  float x = pc.y * 63.0f;
  float x0f = floorf(x), y0f = floorf(y);
  float wx = x - x0f, wy = y - y0f;
  int ix0 = (int)x0f; ix0 = ix0 < 0 ? 0 : (ix0 > 63 ? 63 : ix0);
  int iy0 = (int)y0f; iy0 = iy0 < 0 ? 0 : (iy0 > 63 ? 63 : iy0);
  int ix1 = ix0 + 1 > 63 ? 63 : ix0 + 1;
  int iy1 = iy0 + 1 > 63 ? 63 : iy0 + 1;
  Samp s;
  s.r00 = iy0 * 64 + ix0;
  s.r01 = iy0 * 64 + ix1;
  s.r10 = iy1 * 64 + ix0;
  s.r11 = iy1 * 64 + ix1;
  float w11 = wx * wy;
  s.w11 = w11;
  s.w01 = wx - w11;            // wx*(1-wy)
  s.w10 = wy - w11;            // (1-wx)*wy
  s.w00 = 1.0f - wx - wy + w11;// (1-wx)*(1-wy)
  return s;
}

__global__ __launch_bounds__(BLOCK) void region_pool_kernel(
    const float* __restrict__ fm,      // [B, HW, C]
    const float* __restrict__ coords,  // [B, M, P, 2]
    float* __restrict__ out)           // [B, M, 1, C]
{
  __shared__ __align__(16) float2 lds_pc[Pn];   // 4 KB of point coords

  const int tid = (int)threadIdx.x;
  const int cs  = (int)blockIdx.x & (CSPLIT - 1);
  const int m   = ((int)blockIdx.x >> 2) & (Mn - 1);
  const int b   = (int)blockIdx.x >> 7;         // / (CSPLIT*Mn)

  const float2* gpc = (const float2*)coords + (size_t)(b * Mn + m) * Pn;

  // ---- stage coords into LDS (async DMA path on gfx1250) ----
#if defined(HAVE_ASYNC_LDS)
  {
    const char* gsrc = (const char*)gpc;
    char*       lbas = (char*)lds_pc;
    #pragma unroll
    for (int i = 0; i < (Pn * 8) / (16 * BLOCK); ++i) {   // 4 chunks / thread
      int chunk = tid + i * BLOCK;                         // 16B chunks
      __builtin_amdgcn_global_load_async_to_lds_b128(
          (v4i*)(gsrc + (size_t)chunk * 16),
          (v4i*)(lbas + (size_t)chunk * 16),
          0, 0);
    }
  }
#  if defined(HAVE_WAIT_ASYNC)
  __builtin_amdgcn_s_wait_asynccnt(0);
#  else
  asm volatile("s_wait_asynccnt 0" ::: "memory");
#  endif
#else
  for (int i = tid; i < Pn; i += BLOCK) lds_pc[i] = gpc[i];
#endif
  __syncthreads();

  // ---- gather + weighted accumulate, 2-stage software pipeline ----
  const float4* fm4 = (const float4*)fm;
  const size_t base4 =
      ((size_t)b * HWn * Cn + (size_t)cs * CT) / 4 + (size_t)tid;

  float4 acc; acc.x = 0.f; acc.y = 0.f; acc.z = 0.f; acc.w = 0.f;

  Samp s = make_samp(lds_pc[0]);
  float4 f00 = fm4[base4 + (size_t)s.r00 * ROW4];
  float4 f01 = fm4[base4 + (size_t)s.r01 * ROW4];
  float4 f10 = fm4[base4 + (size_t)s.r10 * ROW4];
  float4 f11 = fm4[base4 + (size_t)s.r11 * ROW4];

  for (int p = 0; p < Pn; ++p) {
    const int pn = (p + 1 < Pn) ? p + 1 : p;
    Samp  sn  = make_samp(lds_pc[pn]);
    float4 n00 = fm4[base4 + (size_t)sn.r00 * ROW4];
    float4 n01 = fm4[base4 + (size_t)sn.r01 * ROW4];
    float4 n10 = fm4[base4 + (size_t)sn.r10 * ROW4];
    float4 n11 = fm4[base4 + (size_t)sn.r11 * ROW4];

    // prefetch rows PFD points ahead (global_prefetch_b8)
    const int pp = p + PFD;
    if (pp < Pn) {
      Samp sp = make_samp(lds_pc[pp]);
      __builtin_prefetch((const void*)(fm4 + base4 + (size_t)sp.r00 * ROW4), 0, 1);
      __builtin_prefetch((const void*)(fm4 + base4 + (size_t)sp.r01 * ROW4), 0, 1);
      __builtin_prefetch((const void*)(fm4 + base4 + (size_t)sp.r10 * ROW4), 0, 1);
      __builtin_prefetch((const void*)(fm4 + base4 + (size_t)sp.r11 * ROW4), 0, 1);
    }

    // accumulate current point while next point's loads are in flight
    acc.x = fmaf(s.w00, f00.x, acc.x);
    acc.y = fmaf(s.w00, f00.y, acc.y);
    acc.z = fmaf(s.w00, f00.z, acc.z);
    acc.w = fmaf(s.w00, f00.w, acc.w);
    acc.x = fmaf(s.w01, f01.x, acc.x);
    acc.y = fmaf(s.w01, f01.y, acc.y);
    acc.z = fmaf(s.w01, f01.z, acc.z);
    acc.w = fmaf(s.w01, f01.w, acc.w);
    acc.x = fmaf(s.w10, f10.x, acc.x);
    acc.y = fmaf(s.w10, f10.y, acc.y);
    acc.z = fmaf(s.w10, f10.z, acc.z);
    acc.w = fmaf(s.w10, f10.w, acc.w);
    acc.x = fmaf(s.w11, f11.x, acc.x);
    acc.y = fmaf(s.w11, f11.y, acc.y);
    acc.z = fmaf(s.w11, f11.z, acc.z);
    acc.w = fmaf(s.w11, f11.w, acc.w);

    s = sn;
    f00 = n00; f01 = n01; f10 = n10; f11 = n11;
  }

  const float inv = 1.0f / (float)Pn;
  acc.x *= inv; acc.y *= inv; acc.z *= inv; acc.w *= inv;

  // out[b, m, 0, cs*CT + 4*tid .. +3]
  const size_t o4 = ((size_t)(b * Mn + m) * Cn + (size_t)cs * CT) / 4 + (size_t)tid;
  ((float4*)out)[o4] = acc;
}

extern "C" void kernel_launch(void* const* d_in, const int* in_sizes, int n_in,
                              void* d_out, int out_size, void* d_ws, size_t ws_size,
                              hipStream_t stream) {
  (void)in_sizes; (void)n_in; (void)d_ws; (void)ws_size; (void)out_size;
  const float* fm     = (const float*)d_in[0];  // [8, 4096, 1024] fp32
  const float* coords = (const float*)d_in[1];  // [8, 32, 512, 2] fp32
  float*       out    = (float*)d_out;          // [8, 32, 1, 1024] fp32

  dim3 grid(Bn * Mn * CSPLIT);   // 1024 blocks
  dim3 block(BLOCK);             // 64 threads (2 waves)
  region_pool_kernel<<<grid, block, 0, stream>>>(fm, coords, out);
}